// SpatiallyVaryingPhysicalLayer_3264175145303
// MI455X (gfx1250) — compile-verified
//
#include <hip/hip_runtime.h>
#include <math.h>
#include <stdint.h>

typedef float v2f __attribute__((ext_vector_type(2)));
typedef float v8f __attribute__((ext_vector_type(8)));

#define NP     49
#define PATCH  128
#define STR    64
#define KS     31
#define NZ     15
#define HID    64
#define IMG    512
#define BATCH  8
#define CH     3
#define NH     7
#define TWO_PI 6.28318530717958647692f

// ---------------------------------------------------------------------------
// Hann window value for 128-length window
__device__ __forceinline__ float hannw(int i) {
    return 0.5f - 0.5f * cosf(TWO_PI * (float)i / 128.0f);
}

// ---------------------------------------------------------------------------
// CDNA5 async memory->LDS copy (GLOBAL_LOAD_ASYNC_TO_LDS_B32, ASYNCcnt).
// ldsaddr = low 32 bits of the generic shared pointer (LDS aperture maps
// addr[31:0] to the LDS offset).
__device__ __forceinline__ void async_load_lds_b32(float* lds_dst, const float* gsrc) {
    unsigned lds = (unsigned)(uintptr_t)lds_dst;
    asm volatile("global_load_async_to_lds_b32 %0, %1, off"
                 :: "v"(lds), "v"(gsrc) : "memory");
}
__device__ __forceinline__ void async_wait_all() {
    asm volatile("s_wait_asynccnt 0x0" ::: "memory");
}

// ---------------------------------------------------------------------------
// One 16x16 output tile of a 32x32 x 32x32 f32 matmul, accumulated into c.
// Uses V_WMMA_F32_16X16X4_F32 chained over 8 k-chunks of 4.
// A,B are row-major 32x32 LDS arrays. asign lets us do the complex subtract.
__device__ __forceinline__ v8f mm32_acc(const float* __restrict__ A,
                                        const float* __restrict__ B,
                                        int tm, int tn, int lane,
                                        v8f c, float asign) {
    const int rowA = tm + (lane & 15);
    const int colB = tn + (lane & 15);
    const int kk   = (lane >> 4) << 1;   // 0 for lanes 0-15, 2 for lanes 16-31
#pragma unroll
    for (int k0 = 0; k0 < 32; k0 += 4) {
        v2f a, b;
        a.x = asign * A[rowA * 32 + k0 + kk];       // VGPR0: K=k0 / K=k0+2
        a.y = asign * A[rowA * 32 + k0 + kk + 1];   // VGPR1: K=k0+1 / K=k0+3
        b.x = B[(k0 + kk) * 32 + colB];
        b.y = B[(k0 + kk + 1) * 32 + colB];
        c = __builtin_amdgcn_wmma_f32_16x16x4_f32(false, a, false, b,
                                                  (short)0, c, false, false);
    }
    return c;
}

// Store a 16x16 f32 accumulator tile to a 32x32 LDS array.
// C/D layout: VGPR r -> row tm+r (lanes 0-15), row tm+r+8 (lanes 16-31).
__device__ __forceinline__ void store_tile(float* __restrict__ D, v8f c,
                                           int tm, int tn, int lane) {
    const int col   = tn + (lane & 15);
    const int rbase = tm + ((lane >> 4) << 3);
#pragma unroll
    for (int r = 0; r < 8; ++r)
        D[(rbase + r) * 32 + col] = c[r];
}

// ---------------------------------------------------------------------------
// Kernel 1: per-position MLP -> Zernike coeffs -> pupil -> centered 31x31 DFT
// (via WMMA f32 matmuls) -> normalized, flipped PSF kernel.
__global__ __launch_bounds__(256) void psf_kernel(
    const float* __restrict__ w1, const float* __restrict__ b1,
    const float* __restrict__ w2, const float* __restrict__ b2,
    const float* __restrict__ w3, const float* __restrict__ b3,
    const float* __restrict__ basis, float* __restrict__ kf_out)
{
    __shared__ float sGr[32 * 32], sGi[32 * 32];
    __shared__ float sPr[32 * 32], sPi[32 * 32];
    __shared__ float sTr[32 * 32], sTi[32 * 32];
    __shared__ float sH1[HID], sH2[HID], sCoef[NZ];
    __shared__ float sRed[256];

    const int tid = threadIdx.x;
    const int n   = blockIdx.x;
    const int iy  = n / NH, ix = n % NH;
    const float gy = ((float)(iy * STR) + 64.0f) / 512.0f * 2.0f - 1.0f;
    const float gx = ((float)(ix * STR) + 64.0f) / 512.0f * 2.0f - 1.0f;

    // --- MLP (2 -> 64 -> 64 -> 15), trivially small ---
    if (tid < HID) {
        float s = fmaf(gy, w1[tid], fmaf(gx, w1[HID + tid], b1[tid]));
        sH1[tid] = fmaxf(s, 0.0f);
    }
    __syncthreads();
    if (tid < HID) {
        float s = b2[tid];
#pragma unroll 8
        for (int k = 0; k < HID; ++k) s = fmaf(sH1[k], w2[k * HID + tid], s);
        sH2[tid] = fmaxf(s, 0.0f);
    }
    __syncthreads();
    if (tid < NZ) {
        float s = b3[tid];
#pragma unroll 8
        for (int k = 0; k < HID; ++k) s = fmaf(sH2[k], w3[k * NZ + tid], s);
        sCoef[tid] = s;
    }
    __syncthreads();

    // --- Shifted DFT matrix G (symmetric), zero-padded to 32x32 ---
    for (int idx = tid; idx < 1024; idx += 256) {
        int u = idx >> 5, x = idx & 31;
        float gr = 0.0f, gi = 0.0f;
        if (u < KS && x < KS) {
            int e = (((u + 16) % 31) * ((x + 16) % 31)) % 31;
            float ang = TWO_PI * (float)e / 31.0f;
            gr = cosf(ang);
            gi = -sinf(ang);
        }
        sGr[idx] = gr; sGi[idx] = gi;
        sPr[idx] = 0.0f; sPi[idx] = 0.0f;
    }
    __syncthreads();

    // --- Pupil: mask * exp(i * phase) ---
    for (int idx = tid; idx < KS * KS; idx += 256) {
        int h = idx / KS, w = idx % KS;
        float ph = 0.0f;
#pragma unroll
        for (int z = 0; z < NZ; ++z) ph = fmaf(sCoef[z], basis[z * KS * KS + idx], ph);
        float gh = -1.0f + (2.0f / 30.0f) * (float)h;
        float gw = -1.0f + (2.0f / 30.0f) * (float)w;
        if (gh * gh + gw * gw <= 1.0f) {
            sPr[h * 32 + w] = cosf(ph);
            sPi[h * 32 + w] = sinf(ph);
        }
    }
    __syncthreads();

    const int lane     = tid & 31;
    const int wave     = tid >> 5;      // 8 waves
    const int imag_out = wave >> 2;     // 0: real part, 1: imag part
    const int tile     = wave & 3;      // 4 16x16 tiles of the 32x32 result
    const int tm       = (tile >> 1) << 4;
    const int tn       = (tile & 1) << 4;

    // --- Stage 1: T = G * P (complex) ---
    {
        v8f c = {0.f, 0.f, 0.f, 0.f, 0.f, 0.f, 0.f, 0.f};
        if (imag_out == 0) {
            c = mm32_acc(sGr, sPr, tm, tn, lane, c, 1.0f);
            c = mm32_acc(sGi, sPi, tm, tn, lane, c, -1.0f);
        } else {
            c = mm32_acc(sGr, sPi, tm, tn, lane, c, 1.0f);
            c = mm32_acc(sGi, sPr, tm, tn, lane, c, 1.0f);
        }
        store_tile(imag_out ? sTi : sTr, c, tm, tn, lane);
    }
    __syncthreads();

    // --- Stage 2: F = T * G  (G symmetric => G^T == G); write into sPr/sPi ---
    {
        v8f c = {0.f, 0.f, 0.f, 0.f, 0.f, 0.f, 0.f, 0.f};
        if (imag_out == 0) {
            c = mm32_acc(sTr, sGr, tm, tn, lane, c, 1.0f);
            c = mm32_acc(sTi, sGi, tm, tn, lane, c, -1.0f);
        } else {
            c = mm32_acc(sTr, sGi, tm, tn, lane, c, 1.0f);
            c = mm32_acc(sTi, sGr, tm, tn, lane, c, 1.0f);
        }
        store_tile(imag_out ? sPi : sPr, c, tm, tn, lane);
    }
    __syncthreads();

    // --- PSF = |F|^2, sum-normalize, flip, store ---
    float lsum = 0.0f;
    for (int idx = tid; idx < KS * KS; idx += 256) {
        int h = idx / KS, w = idx % KS;
        float fr = sPr[h * 32 + w], fi = sPi[h * 32 + w];
        float p = fr * fr + fi * fi;
        sTr[h * 32 + w] = p;    // reuse sTr as psf storage
        lsum += p;
    }
    sRed[tid] = lsum;
    __syncthreads();
    for (int off = 128; off > 0; off >>= 1) {
        if (tid < off) sRed[tid] += sRed[tid + off];
        __syncthreads();
    }
    const float inv = 1.0f / (sRed[0] + 1e-12f);
    for (int idx = tid; idx < KS * KS; idx += 256) {
        int dy = idx / KS, dx = idx % KS;
        kf_out[n * KS * KS + idx] = sTr[(30 - dy) * 32 + (30 - dx)] * inv;
    }
}

// ---------------------------------------------------------------------------
// Kernel 2: depthwise 31x31 conv of zero-padded 128x128 patches with the
// per-position kernel, Hann^2 windowing, scatter-add into accumulator.
// Tile fill uses CDNA5 GLOBAL_LOAD_ASYNC_TO_LDS_B32 (memory -> LDS direct,
// no VGPR staging); patch-boundary zeros are written with regular ds stores
// to disjoint LDS addresses.
#define TILE 64
#define HALO 15
#define LW   (TILE + 2 * HALO)   // 94

__global__ __launch_bounds__(256) void conv_kernel(
    const float* __restrict__ x, const float* __restrict__ kf,
    float* __restrict__ yacc)
{
    __shared__ float sP[LW * LW];
    __shared__ float sK[KS * KS];

    const int tid    = threadIdx.x;
    const int tileId = blockIdx.x;       // 0..3 (2x2 tiles of 64)
    const int c      = blockIdx.y;       // channel
    const int bn     = blockIdx.z;       // 0..391
    const int b      = bn / NP;
    const int n      = bn % NP;
    const int r0     = (n / NH) * STR;
    const int c0     = (n % NH) * STR;
    const int ty0    = (tileId >> 1) * TILE;
    const int tx0    = (tileId & 1) * TILE;

    // kf table: async copy straight into LDS
    for (int i = tid; i < KS * KS; i += 256)
        async_load_lds_b32(&sK[i], &kf[n * KS * KS + i]);

    const float* xs = x + (size_t)(b * CH + c) * IMG * IMG;
    for (int i = tid; i < LW * LW; i += 256) {
        int ly = i / LW, lx = i % LW;
        int py = ty0 - HALO + ly;
        int px = tx0 - HALO + lx;
        // zero-pad at PATCH boundary (matches reference's jnp.pad on patches)
        if (py >= 0 && py < PATCH && px >= 0 && px < PATCH) {
            async_load_lds_b32(&sP[i], &xs[(size_t)(r0 + py) * IMG + (c0 + px)]);
        } else {
            sP[i] = 0.0f;
        }
    }
    async_wait_all();        // s_wait_asynccnt 0
    __syncthreads();

    const int tyq = (tid >> 4) << 2;     // 16x16 threads, 4x4 outputs each
    const int txq = (tid & 15) << 2;

    float acc[4][4];
#pragma unroll
    for (int i = 0; i < 4; ++i)
#pragma unroll
        for (int j = 0; j < 4; ++j) acc[i][j] = 0.0f;

    for (int dy = 0; dy < KS; ++dy) {
        const float* krow = &sK[dy * KS];
#pragma unroll
        for (int i = 0; i < 4; ++i) {
            const float* row = &sP[(tyq + i + dy) * LW + txq];
            float p0 = row[0], p1 = row[1], p2 = row[2];
#pragma unroll
            for (int dx = 0; dx < KS; ++dx) {
                float w  = krow[dx];
                float p3 = row[dx + 3];
                acc[i][0] = fmaf(w, p0, acc[i][0]);
                acc[i][1] = fmaf(w, p1, acc[i][1]);
                acc[i][2] = fmaf(w, p2, acc[i][2]);
                acc[i][3] = fmaf(w, p3, acc[i][3]);
                p0 = p1; p1 = p2; p2 = p3;
            }
        }
    }

    float* ydst = yacc + (size_t)(b * CH + c) * IMG * IMG;
#pragma unroll
    for (int i = 0; i < 4; ++i) {
        int oy   = tyq + i;
        float wy = hannw(ty0 + oy);
#pragma unroll
        for (int j = 0; j < 4; ++j) {
            int ox    = txq + j;
            float val = acc[i][j] * wy * hannw(tx0 + ox);
            atomicAdd(&ydst[(size_t)(r0 + ty0 + oy) * IMG + (c0 + tx0 + ox)], val);
        }
    }
}

// ---------------------------------------------------------------------------
// Kernel 3: accumulate window weights per pixel.
__global__ __launch_bounds__(256) void waccum_kernel(float* __restrict__ wacc)
{
    const int tid = threadIdx.x;
    const int n   = blockIdx.x;
    const int r0  = (n / NH) * STR;
    const int c0  = (n % NH) * STR;
    for (int idx = tid; idx < PATCH * PATCH; idx += 256) {
        int y = idx >> 7, xq = idx & 127;
        atomicAdd(&wacc[(size_t)(r0 + y) * IMG + (c0 + xq)], hannw(y) * hannw(xq));
    }
}

// ---------------------------------------------------------------------------
// Kernel 4: y_hat = y_accum / (w_accum + 1e-8)
__global__ __launch_bounds__(256) void finalize_kernel(
    const float* __restrict__ yacc, const float* __restrict__ wacc,
    float* __restrict__ out)
{
    const size_t idx = (size_t)blockIdx.x * 256 + threadIdx.x;
    const size_t total = (size_t)BATCH * CH * IMG * IMG;
    if (idx < total)
        out[idx] = yacc[idx] / (wacc[idx & (IMG * IMG - 1)] + 1e-8f);
}

// ---------------------------------------------------------------------------
extern "C" void kernel_launch(void* const* d_in, const int* in_sizes, int n_in,
                              void* d_out, int out_size, void* d_ws, size_t ws_size,
                              hipStream_t stream) {
    const float* x     = (const float*)d_in[0];
    const float* w1    = (const float*)d_in[1];
    const float* b1    = (const float*)d_in[2];
    const float* w2    = (const float*)d_in[3];
    const float* b2    = (const float*)d_in[4];
    const float* w3    = (const float*)d_in[5];
    const float* b3    = (const float*)d_in[6];
    const float* basis = (const float*)d_in[7];
    float* out = (float*)d_out;

    float* ws   = (float*)d_ws;
    float* kfp  = ws;                                 // 49*961   = 47089 floats
    float* yacc = ws + 47104;                         // 6291456 floats
    float* wacc = yacc + (size_t)BATCH * CH * IMG * IMG; // 262144 floats

    hipMemsetAsync(yacc, 0, (size_t)BATCH * CH * IMG * IMG * sizeof(float), stream);
    hipMemsetAsync(wacc, 0, (size_t)IMG * IMG * sizeof(float), stream);

    psf_kernel<<<NP, 256, 0, stream>>>(w1, b1, w2, b2, w3, b3, basis, kfp);
    conv_kernel<<<dim3(4, CH, BATCH * NP), 256, 0, stream>>>(x, kfp, yacc);
    waccum_kernel<<<NP, 256, 0, stream>>>(wacc);
    finalize_kernel<<<(BATCH * CH * IMG * IMG + 255) / 256, 256, 0, stream>>>(yacc, wacc, out);
}